// LinSolver_32916629356745
// MI455X (gfx1250) — compile-verified
//
#include <hip/hip_runtime.h>
#include <hip/hip_bf16.h>
#include <math.h>

// Batched primal-dual IPM: one workgroup (128 threads = 4 wave32) per batch
// element. Heavy op per iteration: M = W diag(y/s) W^T (64x64x128 f32 GEMM)
// via V_WMMA_F32_16X16X4_F32 from LDS. Only the LOWER-TRIANGULAR tiles of the
// symmetric M are computed (10 of 16), balanced 3/3/2/2 across the 4 waves.

#define NITERS   30
#define SIGMA_C  0.1f
#define EPS_C    1e-8f
#define STEPF    0.995f

#define MDIM 64
#define NDIM 128
#define LDSW 132   // padded row stride for W / As (132 % 64 = 4 -> conflict free)
#define LDSM 68    // padded row stride for M / L

typedef __attribute__((ext_vector_type(2))) float v2f;
typedef __attribute__((ext_vector_type(8))) float v8f;

struct SolverSmem {
    float W [MDIM * LDSW];   // W row-major (padded)
    float As[MDIM * LDSW];   // W * diag(d)
    float M [MDIM * LDSM];   // normal matrix -> L (lower triangle only used)
    float Ldiag[MDIM];       // Cholesky diagonal (kept separate: no RW races)
    float y[NDIM], s[NDIM], rd[NDIM], rc[NDIM];
    float dsv[NDIM], dyv[NDIM], tv[NDIM];
    float lam[MDIM], rp[MDIM], rhs[MDIM], zv[MDIM], dlam[MDIM];
    float red[4], red2[4];
};

__device__ __forceinline__ float waveSum(float v) {
    #pragma unroll
    for (int o = 16; o > 0; o >>= 1) v += __shfl_xor(v, o, 32);
    return v;
}
__device__ __forceinline__ float waveMin(float v) {
    #pragma unroll
    for (int o = 16; o > 0; o >>= 1) v = fminf(v, __shfl_xor(v, o, 32));
    return v;
}

__global__ __launch_bounds__(128) void ipm_solver_kernel(
    const float* __restrict__ Wg,   // [B,64,128]
    const float* __restrict__ hg,   // [B,64]
    const float* __restrict__ qg,   // [B,128]
    float* __restrict__ out)        // [B,128]  (y)
{
    __shared__ SolverSmem sm;

    const int b    = blockIdx.x;
    const int tid  = threadIdx.x;          // 0..127
    const int lane = tid & 31;
    const int wid  = tid >> 5;             // wave 0..3
    const int mrow = lane & 15;            // M (or N) index within 16-tile
    const int khalf = (lane >> 4) * 2;     // K sub-index select (ISA 7.12.2)

    const float* Wb = Wg + (size_t)b * MDIM * NDIM;

    // ---- stage W into LDS (reused 30x), init state ----
    for (int r = 0; r < MDIM; ++r)
        sm.W[r * LDSW + tid] = Wb[r * NDIM + tid];
    sm.y[tid] = 1.0f;
    sm.s[tid] = 1.0f;
    if (tid < MDIM) sm.lam[tid] = 0.0f;
    __syncthreads();

    // one 16x16 output tile of M = As * W^T (+ eps on diagonal)
    auto do_tile = [&](int it_, int jt_) {
        const int i0 = it_ * 16;
        const int j0 = jt_ * 16;
        v8f acc = {};
        #pragma unroll
        for (int k0 = 0; k0 < NDIM; k0 += 4) {
            v2f a, bm;
            // A 16x4: lane m holds K = khalf, khalf+1
            a[0]  = sm.As[(i0 + mrow) * LDSW + k0 + khalf];
            a[1]  = sm.As[(i0 + mrow) * LDSW + k0 + khalf + 1];
            // B 4x16: B[kk][n] = W[j0+n][k0+kk]
            bm[0] = sm.W [(j0 + mrow) * LDSW + k0 + khalf];
            bm[1] = sm.W [(j0 + mrow) * LDSW + k0 + khalf + 1];
            acc = __builtin_amdgcn_wmma_f32_16x16x4_f32(
                false, a, false, bm, (short)0, acc, false, false);
        }
        // C/D layout: VGPR r -> row i0 + r + 8*(lane>=16), col j0 + (lane&15)
        const int rbase = i0 + ((lane >> 4) ? 8 : 0);
        const int col   = j0 + mrow;
        #pragma unroll
        for (int r = 0; r < 8; ++r) {
            const int row = rbase + r;
            float v = acc[r];
            if (row == col) v += EPS_C;
            sm.M[row * LDSM + col] = v;
        }
    };

    for (int it = 0; it < NITERS; ++it) {
        // ---- mu = sum(y*s)/n  (wave shuffle reduce + tiny cross-wave) ----
        {
            float p = waveSum(sm.y[tid] * sm.s[tid]);
            if (lane == 0) sm.red[wid] = p;
        }
        __syncthreads();
        const float mu = (sm.red[0] + sm.red[1] + sm.red[2] + sm.red[3])
                         * (1.0f / (float)NDIM);

        // ---- r_p = h - W y   (one row per thread, tid<64) ----
        if (tid < MDIM) {
            float acc = hg[(size_t)b * MDIM + tid];
            for (int k = 0; k < NDIM; ++k)
                acc -= sm.W[tid * LDSW + k] * sm.y[k];
            sm.rp[tid] = acc;
        }
        // ---- r_d = q - W^T lam - s; r_c; d = y/s; tv; As = W*diag(d) ----
        {
            float acc = qg[(size_t)b * NDIM + tid];
            for (int i = 0; i < MDIM; ++i)
                acc -= sm.W[i * LDSW + tid] * sm.lam[i];
            const float yv = sm.y[tid], sv = sm.s[tid];
            const float rdv = acc - sv;
            sm.rd[tid] = rdv;
            const float rcv = SIGMA_C * mu - yv * sv;
            sm.rc[tid] = rcv;
            const float dvv = yv / sv;
            sm.tv[tid] = (rcv - yv * rdv) / sv;
            for (int r = 0; r < MDIM; ++r)
                sm.As[r * LDSW + tid] = sm.W[r * LDSW + tid] * dvv;
        }
        __syncthreads();

        // ---- lower triangle of symmetric M via WMMA; 10 tiles, 3/3/2/2 ----
        // (wave-uniform branches: EXEC all-ones around every WMMA)
        if (wid == 0)      { do_tile(0,0); do_tile(2,0); do_tile(3,0); }
        else if (wid == 1) { do_tile(1,0); do_tile(2,1); do_tile(3,1); }
        else if (wid == 2) { do_tile(1,1); do_tile(3,2); }
        else               { do_tile(2,2); do_tile(3,3); }

        // ---- rhs = r_p - W * tv ----
        if (tid < MDIM) {
            float acc = sm.rp[tid];
            for (int k = 0; k < NDIM; ++k)
                acc -= sm.W[tid * LDSW + k] * sm.tv[k];
            sm.rhs[tid] = acc;
        }
        __syncthreads();

        // ---- in-place Cholesky (lower L; diagonal kept in Ldiag) ----
        for (int k = 0; k < MDIM; ++k) {
            const float dkk = sm.M[k * LDSM + k];   // not written this phase
            const float sq  = sqrtf(dkk);
            if (tid == k) sm.Ldiag[k] = sq;
            if (tid > k && tid < MDIM)
                sm.M[tid * LDSM + k] *= (1.0f / sq);
            __syncthreads();
            if (tid > k && tid < MDIM) {
                const float lik = sm.M[tid * LDSM + k];   // col k: read-only now
                for (int j = k + 1; j <= tid; ++j)
                    sm.M[tid * LDSM + j] -= lik * sm.M[j * LDSM + k];
            }
            __syncthreads();
        }

        // ---- forward solve L z = rhs: residual in rhs, solution -> zv ----
        for (int i = 0; i < MDIM; ++i) {
            const float zi = sm.rhs[i] / sm.Ldiag[i];  // rhs[i] stable this step
            if (tid == i) sm.zv[i] = zi;
            if (tid > i && tid < MDIM)
                sm.rhs[tid] -= sm.M[tid * LDSM + i] * zi;
            __syncthreads();
        }
        // ---- backward solve L^T dlam = z: residual in zv, solution -> dlam --
        for (int i = MDIM - 1; i >= 0; --i) {
            const float xi = sm.zv[i] / sm.Ldiag[i];   // zv[i] stable this step
            if (tid == i) sm.dlam[i] = xi;
            if (tid < i)
                sm.zv[tid] -= sm.M[i * LDSM + tid] * xi;
            __syncthreads();
        }

        // ---- ds = r_d - W^T dlam ; dy = (r_c - y*ds)/s ; step ratios ----
        {
            float acc = sm.rd[tid];
            for (int i = 0; i < MDIM; ++i)
                acc -= sm.W[i * LDSW + tid] * sm.dlam[i];
            const float dsv = acc;
            sm.dsv[tid] = dsv;
            const float dyv = (sm.rc[tid] - sm.y[tid] * dsv) / sm.s[tid];
            sm.dyv[tid] = dyv;
            const float BIG = 3.0e38f;
            const float rpw = waveMin((dyv < 0.0f) ? (-sm.y[tid] / dyv) : BIG);
            const float rdw = waveMin((dsv < 0.0f) ? (-sm.s[tid] / dsv) : BIG);
            if (lane == 0) { sm.red[wid] = rpw; sm.red2[wid] = rdw; }
        }
        __syncthreads();
        const float ap = fminf(1.0f, STEPF * fminf(fminf(sm.red [0], sm.red [1]),
                                                   fminf(sm.red [2], sm.red [3])));
        const float ad = fminf(1.0f, STEPF * fminf(fminf(sm.red2[0], sm.red2[1]),
                                                   fminf(sm.red2[2], sm.red2[3])));

        // ---- update (each thread touches only its own elements) ----
        sm.y[tid] += ap * sm.dyv[tid];
        sm.s[tid] += ad * sm.dsv[tid];
        if (tid < MDIM) sm.lam[tid] += ad * sm.dlam[tid];
        __syncthreads();
    }

    out[(size_t)b * NDIM + tid] = sm.y[tid];
}

extern "C" void kernel_launch(void* const* d_in, const int* in_sizes, int n_in,
                              void* d_out, int out_size, void* d_ws, size_t ws_size,
                              hipStream_t stream) {
    const float* W = (const float*)d_in[0];   // [B,64,128]
    const float* h = (const float*)d_in[1];   // [B,64]
    const float* q = (const float*)d_in[2];   // [B,128]
    float* out = (float*)d_out;               // [B,128]

    const int B = in_sizes[1] / MDIM;         // h has B*64 elements
    ipm_solver_kernel<<<dim3(B), dim3(128), 0, stream>>>(W, h, q, out);
}